// multi_head_attn_12335146074310
// MI455X (gfx1250) — compile-verified
//
#include <hip/hip_runtime.h>
#include <hip/hip_bf16.h>

// ---------------------------------------------------------------------------
// Fused multi-head spatial attention for MI455X (gfx1250, wave32, WMMA).
//
//   prep_kernel : Qs, K (64x9216) elementwise, stored FEATURE-PAIR
//                 INTERLEAVED so WMMA A/B operands load as single b64s.
//   conv_kernel : V (32x9216) = affine(3x3 conv of X).
//   attn_kernel : flash-style softmax(Qs^T K) V^T, 64-column K-tiles,
//                 96 v_wmma_f32_16x16x4_f32 per iteration, sim never in HBM.
// ---------------------------------------------------------------------------

typedef __attribute__((ext_vector_type(2))) float v2f;
typedef __attribute__((ext_vector_type(8))) float v8f;

#define N_HEADS 8
#define NF      64
#define NJ      (NF / 2)      // feature pairs
#define CD      32
#define IMG     96
#define PIX     (IMG * IMG)   // 9216 = 144 * 64, no tail handling needed
#define KT      64            // K-tile width per flash iteration

// ---------------------------------------------------------------------------
// Pair-interleaved prep:  Qs2[j][p] = {Qs[2j][p], Qs[2j+1][p]},  same for K2.
// This matches the 16x16x4 WMMA per-lane operand layout exactly.
// ---------------------------------------------------------------------------
__global__ void prep_kernel(const float* __restrict__ X,
                            const float* __restrict__ WQ_task,
                            const float* __restrict__ BQ_task,
                            const float* __restrict__ WK_task,
                            const float* __restrict__ BK_task,
                            const float* __restrict__ WQ_t1,
                            const float* __restrict__ WQ_x,
                            const float* __restrict__ BQ,
                            const float* __restrict__ WK_x,
                            const float* __restrict__ BK,
                            const float* __restrict__ prevQ,
                            v2f* __restrict__ Qs2,
                            v2f* __restrict__ K2) {
    int idx = blockIdx.x * blockDim.x + threadIdx.x;   // j*PIX + p
    if (idx >= NJ * PIX) return;
    int j = idx / PIX;
    int p = idx - j * PIX;
    int f0 = 2 * j, f1 = 2 * j + 1;

    float x0 = X[f0 * PIX + p];
    float x1 = X[f1 * PIX + p];

    v2f kv;
    {
        float k0 = WK_x[f0] * x0 + BK[f0];
        float k1 = WK_x[f1] * x1 + BK[f1];
        k0 = k0 > 0.f ? k0 : 0.f;
        k1 = k1 > 0.f ? k1 : 0.f;
        kv.x = WK_task[f0] * k0 + BK_task[f0];
        kv.y = WK_task[f1] * k1 + BK_task[f1];
    }
    K2[idx] = kv;

    float wqx0 = WQ_x[f0] * x0;
    float wqx1 = WQ_x[f1] * x1;
    v2f qv = {};
#pragma unroll
    for (int a = 0; a < N_HEADS; ++a) {
        int af0 = a * NF + f0, af1 = a * NF + f1;
        float t0 = WQ_t1[af0] * prevQ[(size_t)af0 * PIX + p] + wqx0 + BQ[af0];
        float t1 = WQ_t1[af1] * prevQ[(size_t)af1 * PIX + p] + wqx1 + BQ[af1];
        t0 = t0 > 0.f ? t0 : 0.f;
        t1 = t1 > 0.f ? t1 : 0.f;
        qv.x += WQ_task[af0] * t0 + BQ_task[af0];
        qv.y += WQ_task[af1] * t1 + BQ_task[af1];
    }
    Qs2[idx] = qv;
}

// ---------------------------------------------------------------------------
// V[c,p] = WV_task[c] * (conv3x3(X)[c,p] + Vconv_b[c]) + BV_task[c]
// ---------------------------------------------------------------------------
__global__ void conv_kernel(const float* __restrict__ X,
                            const float* __restrict__ Wc,
                            const float* __restrict__ bc,
                            const float* __restrict__ WV_task,
                            const float* __restrict__ BV_task,
                            float* __restrict__ Vm) {
    int idx = blockIdx.x * blockDim.x + threadIdx.x;   // c*PIX + p
    if (idx >= CD * PIX) return;
    int c = idx / PIX;
    int p = idx - c * PIX;
    int y = p / IMG;
    int x = p - y * IMG;

    float acc = 0.f;
    for (int f = 0; f < NF; ++f) {
        const float* wr = &Wc[(c * NF + f) * 9];
        const float* xr = &X[f * PIX];
#pragma unroll
        for (int ky = 0; ky < 3; ++ky) {
            int yy = y + ky - 1;
            if (yy < 0 || yy >= IMG) continue;
#pragma unroll
            for (int kx = 0; kx < 3; ++kx) {
                int xx = x + kx - 1;
                if (xx < 0 || xx >= IMG) continue;
                acc += wr[ky * 3 + kx] * xr[yy * IMG + xx];
            }
        }
    }
    acc += bc[c];
    Vm[idx] = WV_task[c] * acc + BV_task[c];
}

// ---------------------------------------------------------------------------
// Flash attention: one wave owns 16 query rows; 64-column K-tiles.
//   S (16x64)  : 64 x V_WMMA_F32_16X16X4_F32   (contract over 64 features)
//   softmax    : VALU-combine 4 sub-tiles, then ONE 4-step shfl_xor per row
//   O += P V^T : 32 x WMMA (K-dim 64, two 16-channel chunks), P via 4KB LDS
// ---------------------------------------------------------------------------
__global__ __launch_bounds__(32) void attn_kernel(const v2f* __restrict__ Qs2,
                                                  const v2f* __restrict__ K2,
                                                  const float* __restrict__ Vm,
                                                  float* __restrict__ out) {
    __shared__ float pLDS[16 * KT];

    const int lane  = threadIdx.x;
    const int m     = lane & 15;   // A-row / B-col / C-col index
    const int h     = lane >> 4;   // half-wave select
    const int qbase = blockIdx.x * 16;

    // A = Qs^T[16 rows x 64 features], pair-interleaved -> one b64 per step.
    v2f aQ[16];
#pragma unroll
    for (int t = 0; t < 16; ++t)
        aQ[t] = Qs2[(2 * t + h) * PIX + qbase + m];

    v8f acc0 = {};            // O[q][c], c=0..15  (C/D layout: row r+8h, col m)
    v8f acc1 = {};            // O[q][c], c=16..31
    float mrow[8], lrow[8];
#pragma unroll
    for (int r = 0; r < 8; ++r) { mrow[r] = -3.0e38f; lrow[r] = 0.f; }

    for (int kt = 0; kt < PIX / KT; ++kt) {
        const int kbase = kt * KT;

        // Prefetch next iteration's K/V blocks (all L2-resident; hides L2 lat).
        if (kt + 1 < PIX / KT) {
            __builtin_prefetch(&K2[(h)*PIX + kbase + KT + m], 0, 3);
            __builtin_prefetch(&Vm[(m)*PIX + kbase + KT + 2 * h], 0, 3);
        }

        // ---- S = Qs^T @ K : four 16x16 sub-tiles, K-dim 64 ----
        v8f s0 = {}, s1 = {}, s2 = {}, s3 = {};
#pragma unroll
        for (int t = 0; t < 16; ++t) {
            const v2f* kp = &K2[(2 * t + h) * PIX + kbase + m];
            s0 = __builtin_amdgcn_wmma_f32_16x16x4_f32(false, aQ[t], false, kp[0],  (short)0, s0, false, false);
            s1 = __builtin_amdgcn_wmma_f32_16x16x4_f32(false, aQ[t], false, kp[16], (short)0, s1, false, false);
            s2 = __builtin_amdgcn_wmma_f32_16x16x4_f32(false, aQ[t], false, kp[32], (short)0, s2, false, false);
            s3 = __builtin_amdgcn_wmma_f32_16x16x4_f32(false, aQ[t], false, kp[48], (short)0, s3, false, false);
        }

        // ---- online softmax over 64 columns (rows q = r + 8h) ----
#pragma unroll
        for (int r = 0; r < 8; ++r) {
            float v0 = s0[r], v1 = s1[r], v2 = s2[r], v3 = s3[r];
            float rmax = fmaxf(fmaxf(v0, v1), fmaxf(v2, v3));
            rmax = fmaxf(rmax, __shfl_xor(rmax, 1));
            rmax = fmaxf(rmax, __shfl_xor(rmax, 2));
            rmax = fmaxf(rmax, __shfl_xor(rmax, 4));
            rmax = fmaxf(rmax, __shfl_xor(rmax, 8));
            float mnew  = fmaxf(mrow[r], rmax);
            float scale = __expf(mrow[r] - mnew);
            float p0 = __expf(v0 - mnew);
            float p1 = __expf(v1 - mnew);
            float p2 = __expf(v2 - mnew);
            float p3 = __expf(v3 - mnew);
            float rs = (p0 + p1) + (p2 + p3);
            rs += __shfl_xor(rs, 1);
            rs += __shfl_xor(rs, 2);
            rs += __shfl_xor(rs, 4);
            rs += __shfl_xor(rs, 8);
            lrow[r] = lrow[r] * scale + rs;
            mrow[r] = mnew;
            acc0[r] *= scale;
            acc1[r] *= scale;
            int row = (r + 8 * h) * KT + m;       // C/D layout -> row-major P
            pLDS[row +  0] = p0;
            pLDS[row + 16] = p1;
            pLDS[row + 32] = p2;
            pLDS[row + 48] = p3;
        }
        __syncthreads();

        // ---- O += P @ V^T, K-dim 64, two 16-channel chunks ----
#pragma unroll
        for (int t = 0; t < 16; ++t) {
            int k0 = 4 * t + 2 * h;
            v2f a  = *(const v2f*)&pLDS[m * KT + k0];                 // ds b64
            v2f b0 = *(const v2f*)&Vm[(0 * 16 + m) * PIX + kbase + k0]; // b64
            v2f b1 = *(const v2f*)&Vm[(1 * 16 + m) * PIX + kbase + k0]; // b64
            acc0 = __builtin_amdgcn_wmma_f32_16x16x4_f32(false, a, false, b0, (short)0, acc0, false, false);
            acc1 = __builtin_amdgcn_wmma_f32_16x16x4_f32(false, a, false, b1, (short)0, acc1, false, false);
        }
        __syncthreads();
    }

    // ---- normalize and write out[c][q] ----
#pragma unroll
    for (int r = 0; r < 8; ++r) {
        int q = qbase + r + 8 * h;
        float inv = 1.0f / lrow[r];
        out[(0 * 16 + m) * PIX + q] = acc0[r] * inv;
        out[(1 * 16 + m) * PIX + q] = acc1[r] * inv;
    }
}

// ---------------------------------------------------------------------------
extern "C" void kernel_launch(void* const* d_in, const int* in_sizes, int n_in,
                              void* d_out, int out_size, void* d_ws, size_t ws_size,
                              hipStream_t stream) {
    (void)in_sizes; (void)n_in; (void)out_size; (void)ws_size;

    const float* X       = (const float*)d_in[0];
    const float* WQ_task = (const float*)d_in[1];
    const float* BQ_task = (const float*)d_in[2];
    const float* WK_task = (const float*)d_in[3];
    const float* BK_task = (const float*)d_in[4];
    const float* WV_task = (const float*)d_in[5];
    const float* BV_task = (const float*)d_in[6];
    const float* WQ_t1   = (const float*)d_in[7];
    const float* WQ_x    = (const float*)d_in[8];
    const float* BQ      = (const float*)d_in[9];
    const float* WK_x    = (const float*)d_in[10];
    const float* BK      = (const float*)d_in[11];
    const float* Vconv_w = (const float*)d_in[12];
    const float* Vconv_b = (const float*)d_in[13];
    const float* prevQ   = (const float*)d_in[14];

    float* ws = (float*)d_ws;                       // 5.9 MB fp32 scratch
    v2f*   Qs2 = (v2f*)ws;                          // (32, 9216) pairs
    v2f*   K2  = (v2f*)(ws + (size_t)NF * PIX);     // (32, 9216) pairs
    float* Vm  = ws + (size_t)2 * NF * PIX;         // (32, 9216)

    prep_kernel<<<(NJ * PIX + 255) / 256, 256, 0, stream>>>(
        X, WQ_task, BQ_task, WK_task, BK_task, WQ_t1, WQ_x, BQ, WK_x, BK,
        prevQ, Qs2, K2);

    conv_kernel<<<(CD * PIX + 255) / 256, 256, 0, stream>>>(
        X, Vconv_w, Vconv_b, WV_task, BV_task, Vm);

    attn_kernel<<<PIX / 16, 32, 0, stream>>>(Qs2, K2, Vm, (float*)d_out);
}